// _PointNet2SetAbstractionMSG_8246337208757
// MI455X (gfx1250) — compile-verified
//
#include <hip/hip_runtime.h>

#define B_  16
#define N_  16384
#define S_  1024
#define CF_ 16

typedef __attribute__((ext_vector_type(2))) float v2f;
typedef __attribute__((ext_vector_type(8))) float v8f;

struct LayerP { const float *W, *b, *ga, *be, *mu, *va; };
struct AllP   { LayerP l[6]; };

// ---------------------------------------------------------------------------
// 0) Pre-swizzle weights into B-operand lane order + fold BN into scale/shift.
//    Layout per layer: [swW: kts*nts*64] [scale: C] [shift: C]
//    swW index = (kt*nts + nt)*64 + lane*2 + v, value = W[n][k] (k = kt*4+v+2*(lane>=16))
// ---------------------------------------------------------------------------
__global__ void swizzle_weights_kernel(AllP P, float* __restrict__ wsw) {
  const int inr[6]  = {19, 32, 32, 19, 64, 64};
  const int kts[6]  = { 5,  8,  8,  5, 16, 16};
  const int outc[6] = {32, 32, 64, 64, 64,128};
  int off = 0;
  const int lI = blockIdx.x;
  for (int i = 0; i < 6; ++i) {
    if (i == lI) break;
    off += kts[i] * (outc[i] / 16) * 64 + 2 * outc[i];
  }
  const LayerP p = P.l[lI];
  const int nts = outc[lI] / 16;
  const int nW  = kts[lI] * nts * 64;
  float* dstW = wsw + off;
  for (int t = threadIdx.x; t < nW; t += blockDim.x) {
    int tile = t >> 6, e = t & 63;
    int L = e >> 1, v = e & 1;
    int kt = tile / nts, nt = tile % nts;
    int k = kt * 4 + v + ((L >= 16) ? 2 : 0);
    int n = nt * 16 + (L & 15);
    dstW[t] = (k < inr[lI]) ? p.W[n * inr[lI] + k] : 0.f;
  }
  float* sc = dstW + nW;
  float* sh = sc + outc[lI];
  for (int o = threadIdx.x; o < outc[lI]; o += blockDim.x) {
    float af = p.ga[o] * rsqrtf(p.va[o] + 1e-5f);
    sc[o] = af;
    sh[o] = (p.b[o] - p.mu[o]) * af + p.be[o];
  }
}

// ---------------------------------------------------------------------------
// 1) Furthest point sampling: one 1024-thread block per batch.
//    Each thread owns 16 points (xyz + running min-dist in registers).
// ---------------------------------------------------------------------------
__global__ __launch_bounds__(1024) void fps_kernel(const float* __restrict__ xyz,
                                                   float* __restrict__ cents,
                                                   float* __restrict__ out_c) {
  const int b = blockIdx.x;
  const int t = threadIdx.x;
  const float* xb = xyz + (size_t)b * N_ * 3;
  float px[16], py[16], pz[16], dd[16];
#pragma unroll
  for (int j = 0; j < 16; ++j) {
    int pidx = t + j * 1024;
    const float* pp = xb + (size_t)pidx * 3;
    px[j] = pp[0]; py[j] = pp[1]; pz[j] = pp[2];
    dd[j] = 1e10f;
  }
  __shared__ float scx, scy, scz;
  __shared__ int   sidx;
  __shared__ float rd[32];
  __shared__ int   ri[32];
  if (t == 0) sidx = 0;
  __syncthreads();

  for (int s = 0; s < S_; ++s) {
    const int far = sidx;
    if (t == (far & 1023)) {
      const int jj = far >> 10;
      float ox = 0.f, oy = 0.f, oz = 0.f;
#pragma unroll
      for (int j = 0; j < 16; ++j)
        if (j == jj) { ox = px[j]; oy = py[j]; oz = pz[j]; }
      scx = ox; scy = oy; scz = oz;
      float* o1 = cents + ((size_t)b * S_ + s) * 3;
      o1[0] = ox; o1[1] = oy; o1[2] = oz;
      float* o2 = out_c + ((size_t)b * S_ + s) * 3;
      o2[0] = ox; o2[1] = oy; o2[2] = oz;
    }
    __syncthreads();
    const float cx = scx, cy = scy, cz = scz;
    float bd = -1.f; int bi = 0;
#pragma unroll
    for (int j = 0; j < 16; ++j) {
      float dx = px[j] - cx, dy = py[j] - cy, dz = pz[j] - cz;
      float d2 = dx * dx + dy * dy + dz * dz;
      float nd = fminf(dd[j], d2);
      dd[j] = nd;
      if (nd > bd) { bd = nd; bi = t + j * 1024; }
    }
#pragma unroll
    for (int off = 16; off > 0; off >>= 1) {
      float od = __shfl_down(bd, off, 32);
      int   oi = __shfl_down(bi, off, 32);
      if (od > bd || (od == bd && oi < bi)) { bd = od; bi = oi; }
    }
    if ((t & 31) == 0) { rd[t >> 5] = bd; ri[t >> 5] = bi; }
    __syncthreads();
    if (t < 32) {
      bd = rd[t]; bi = ri[t];
#pragma unroll
      for (int off = 16; off > 0; off >>= 1) {
        float od = __shfl_down(bd, off, 32);
        int   oi = __shfl_down(bi, off, 32);
        if (od > bd || (od == bd && oi < bi)) { bd = od; bi = oi; }
      }
      if (t == 0) sidx = bi;
    }
    __syncthreads();
  }
}

// ---------------------------------------------------------------------------
// 2) Ball query (both scales in one scan): one wave per centroid.
//    First-K by ascending index via ballot + popcount rank; pad with first hit.
// ---------------------------------------------------------------------------
__global__ __launch_bounds__(256) void ballquery_kernel(const float* __restrict__ xyz,
                                                        const float* __restrict__ cents,
                                                        int* __restrict__ gi0,
                                                        int* __restrict__ gi1) {
  const int wid = threadIdx.x >> 5;
  const int L   = threadIdx.x & 31;
  const int g   = blockIdx.x * 8 + wid;
  const int b   = g >> 10;
  const float* xb = xyz + (size_t)b * N_ * 3;
  const float cx = cents[g * 3 + 0], cy = cents[g * 3 + 1], cz = cents[g * 3 + 2];
  const float R0 = 0.04f, R1 = 0.16f;
  int cnt0 = 0, cnt1 = 0, f0 = 0, f1 = 0;
  const unsigned lt = (1u << L) - 1u;
  for (int base = 0; base < N_ && (cnt0 < 32 || cnt1 < 64); base += 32) {
    const int i = base + L;
    const float* pp = xb + (size_t)i * 3;
    float dx = pp[0] - cx, dy = pp[1] - cy, dz = pp[2] - cz;
    float d2 = dx * dx + dy * dy + dz * dz;
    unsigned m0 = __builtin_amdgcn_ballot_w32(d2 <= R0);
    unsigned m1 = __builtin_amdgcn_ballot_w32(d2 <= R1);
    if (m0) {
      if (cnt0 == 0) f0 = base + __builtin_ctz(m0);
      int slot = cnt0 + __builtin_popcount(m0 & lt);
      if ((d2 <= R0) && slot < 32) gi0[(size_t)g * 32 + slot] = i;
      cnt0 += __builtin_popcount(m0);
    }
    if (m1) {
      if (cnt1 == 0) f1 = base + __builtin_ctz(m1);
      int slot = cnt1 + __builtin_popcount(m1 & lt);
      if ((d2 <= R1) && slot < 64) gi1[(size_t)g * 64 + slot] = i;
      cnt1 += __builtin_popcount(m1);
    }
  }
  cnt0 = cnt0 < 32 ? cnt0 : 32;
  cnt1 = cnt1 < 64 ? cnt1 : 64;
  for (int slot = cnt0 + L; slot < 32; slot += 32) gi0[(size_t)g * 32 + slot] = f0;
  for (int slot = cnt1 + L; slot < 64; slot += 32) gi1[(size_t)g * 64 + slot] = f1;
}

// ---------------------------------------------------------------------------
// 3) Grouping + MLP (WMMA f32 16x16x4) + max-pool.
// ---------------------------------------------------------------------------
template<int MTS, int KTS, int NTS, int SA, int SD>
__device__ __forceinline__ void mlp_layer(const float* __restrict__ A,
                                          const float* __restrict__ gW,
                                          const float* __restrict__ gSc,
                                          const float* __restrict__ gSh,
                                          float* __restrict__ D, int L) {
  const int hi2 = (L >= 16) ? 2 : 0;
  const int hi8 = (L >= 16) ? 8 : 0;
  const int li  = L & 15;
#pragma unroll
  for (int nt = 0; nt < NTS; ++nt) {
    v2f breg[KTS];
#pragma unroll
    for (int kt = 0; kt < KTS; ++kt)
      breg[kt] = *(const v2f*)(gW + (kt * NTS + nt) * 64 + L * 2);
    const float scv = gSc[nt * 16 + li];
    const float shv = gSh[nt * 16 + li];
#pragma unroll
    for (int mt = 0; mt < MTS; ++mt) {
      v8f acc = {0.f, 0.f, 0.f, 0.f, 0.f, 0.f, 0.f, 0.f};
#pragma unroll
      for (int kt = 0; kt < KTS; ++kt) {
        v2f a = *(const v2f*)(A + (mt * 16 + li) * SA + kt * 4 + hi2);
        acc = __builtin_amdgcn_wmma_f32_16x16x4_f32(false, a, false, breg[kt],
                                                    (short)0, acc, false, false);
      }
#pragma unroll
      for (int r = 0; r < 8; ++r) {
        float v = fmaxf(acc[r] * scv + shv, 0.f);
        D[(mt * 16 + r + hi8) * SD + nt * 16 + li] = v;
      }
    }
  }
}

template<int MTS, int KTS, int NTS, int SA>
__device__ __forceinline__ void mlp_layer_maxout(const float* __restrict__ A,
                                                 const float* __restrict__ gW,
                                                 const float* __restrict__ gSc,
                                                 const float* __restrict__ gSh,
                                                 float* __restrict__ outp, int L) {
  const int hi2 = (L >= 16) ? 2 : 0;
  const int li  = L & 15;
#pragma unroll
  for (int nt = 0; nt < NTS; ++nt) {
    v2f breg[KTS];
#pragma unroll
    for (int kt = 0; kt < KTS; ++kt)
      breg[kt] = *(const v2f*)(gW + (kt * NTS + nt) * 64 + L * 2);
    const float scv = gSc[nt * 16 + li];
    const float shv = gSh[nt * 16 + li];
    float m = 0.f;
#pragma unroll
    for (int mt = 0; mt < MTS; ++mt) {
      v8f acc = {0.f, 0.f, 0.f, 0.f, 0.f, 0.f, 0.f, 0.f};
#pragma unroll
      for (int kt = 0; kt < KTS; ++kt) {
        v2f a = *(const v2f*)(A + (mt * 16 + li) * SA + kt * 4 + hi2);
        acc = __builtin_amdgcn_wmma_f32_16x16x4_f32(false, a, false, breg[kt],
                                                    (short)0, acc, false, false);
      }
#pragma unroll
      for (int r = 0; r < 8; ++r)
        m = fmaxf(m, fmaxf(acc[r] * scv + shv, 0.f));
    }
    m = fmaxf(m, __shfl_xor(m, 16, 32));
    if (L < 16) outp[nt * 16 + li] = m;
  }
}

template<int K, int C1, int C2, int C3, int WAVES, int OUT_OFF>
__global__ __launch_bounds__(WAVES * 32) void group_mlp_kernel(
    const float* __restrict__ xyz, const float* __restrict__ feat,
    const int* __restrict__ gi, const float* __restrict__ cents,
    const float* __restrict__ wsw, int offW1, int offW2, int offW3,
    float* __restrict__ outf) {
  static_assert(C1 == C2, "ping-pong buffers share stride");
  constexpr int MTS = K / 16;
  constexpr int SX  = 20;          // 19 channels + 1 zero pad, stride % 64 == 20 (conflict-free)
  constexpr int SH  = C1 + 4;      // stride % 32 == 4 -> conflict-free A loads / D stores
  __shared__ __align__(16) float sX [WAVES][K * SX];
  __shared__ __align__(16) float sH1[WAVES][K * SH];
  __shared__ __align__(16) float sH2[WAVES][K * SH];

  const int wid = threadIdx.x >> 5;
  const int L   = threadIdx.x & 31;
  const int g   = blockIdx.x * WAVES + wid;
  const int b   = g >> 10;

  const float* xb = xyz + (size_t)b * N_ * 3;
  const float* fb = feat + (size_t)b * N_ * CF_;
  const float cx = cents[g * 3 + 0], cy = cents[g * 3 + 1], cz = cents[g * 3 + 2];
  float* Xw = sX[wid];
#pragma unroll
  for (int r = 0; r < K / 32; ++r) {
    int j = L + r * 32;
    int idx = gi[(size_t)g * K + j];
    idx = (idx < N_) ? idx : (N_ - 1);
    const float* pp = xb + (size_t)idx * 3;
    float* xr = Xw + j * SX;
    xr[0] = pp[0] - cx; xr[1] = pp[1] - cy; xr[2] = pp[2] - cz;
    const float4* fp = (const float4*)(fb + (size_t)idx * CF_);
    float4 q0 = fp[0], q1 = fp[1], q2 = fp[2], q3 = fp[3];
    xr[ 3] = q0.x; xr[ 4] = q0.y; xr[ 5] = q0.z; xr[ 6] = q0.w;
    xr[ 7] = q1.x; xr[ 8] = q1.y; xr[ 9] = q1.z; xr[10] = q1.w;
    xr[11] = q2.x; xr[12] = q2.y; xr[13] = q2.z; xr[14] = q2.w;
    xr[15] = q3.x; xr[16] = q3.y; xr[17] = q3.z; xr[18] = q3.w;
    xr[19] = 0.f;
  }
  __syncthreads();

  const float* W1  = wsw + offW1;
  const float* Sc1 = W1 + 5 * (C1 / 16) * 64;        const float* Sh1 = Sc1 + C1;
  const float* W2  = wsw + offW2;
  const float* Sc2 = W2 + (C1 / 4) * (C2 / 16) * 64; const float* Sh2 = Sc2 + C2;
  const float* W3  = wsw + offW3;
  const float* Sc3 = W3 + (C2 / 4) * (C3 / 16) * 64; const float* Sh3 = Sc3 + C3;

  mlp_layer<MTS, 5,      C1 / 16, SX, SH>(Xw,       W1, Sc1, Sh1, sH1[wid], L);
  __syncthreads();
  mlp_layer<MTS, C1 / 4, C2 / 16, SH, SH>(sH1[wid], W2, Sc2, Sh2, sH2[wid], L);
  __syncthreads();
  mlp_layer_maxout<MTS, C2 / 4, C3 / 16, SH>(sH2[wid], W3, Sc3, Sh3,
                                             outf + (size_t)g * 192 + OUT_OFF, L);
}

// ---------------------------------------------------------------------------
extern "C" void kernel_launch(void* const* d_in, const int* in_sizes, int n_in,
                              void* d_out, int out_size, void* d_ws, size_t ws_size,
                              hipStream_t stream) {
  (void)in_sizes; (void)n_in; (void)out_size; (void)ws_size;
  const float* xyz  = (const float*)d_in[0];
  const float* feat = (const float*)d_in[1];
  AllP P;
  for (int s = 0; s < 2; ++s)
    for (int l = 0; l < 3; ++l) {
      int base = 2 + s * 18 + l * 6;
      LayerP& q = P.l[s * 3 + l];
      q.W  = (const float*)d_in[base + 0];
      q.b  = (const float*)d_in[base + 1];
      q.ga = (const float*)d_in[base + 2];
      q.be = (const float*)d_in[base + 3];
      q.mu = (const float*)d_in[base + 4];
      q.va = (const float*)d_in[base + 5];
    }

  float* ws    = (float*)d_ws;
  float* cents = ws;                       // B*S*3 = 49152 floats
  float* wsw   = ws + 49152;               // swizzled weights + affine: 18048 floats
  int*   gi0   = (int*)(ws + 67200);       // B*S*32 ints
  int*   gi1   = gi0 + (size_t)B_ * S_ * 32;

  float* out_c = (float*)d_out;            // centroids (B,S,3)
  float* out_f = out_c + (size_t)B_ * S_ * 3;  // features (B,S,192)

  swizzle_weights_kernel<<<6, 256, 0, stream>>>(P, wsw);
  fps_kernel<<<B_, 1024, 0, stream>>>(xyz, cents, out_c);
  ballquery_kernel<<<(B_ * S_) / 8, 256, 0, stream>>>(xyz, cents, gi0, gi1);
  // scale 0: K=32, 19->32->32->64, out cols [0,64)
  group_mlp_kernel<32, 32, 32, 64, 8, 0>
      <<<(B_ * S_) / 8, 8 * 32, 0, stream>>>(xyz, feat, gi0, cents, wsw, 0, 704, 1792, out_f);
  // scale 1: K=64, 19->64->64->128, out cols [64,192)
  group_mlp_kernel<64, 64, 64, 128, 4, 64>
      <<<(B_ * S_) / 4, 4 * 32, 0, stream>>>(xyz, feat, gi1, cents, wsw, 3968, 5376, 9600, out_f);
}